// ODConv1x1Linear_43937515438472
// MI455X (gfx1250) — compile-verified
//
#include <hip/hip_runtime.h>
#include <math.h>

// Problem constants (from reference)
#define B_    32
#define CIN   512
#define COUT  512
#define HW    3136     // 56*56
#define KNUM  4
#define ATT   32
#define EPS_  1e-5f

typedef __attribute__((ext_vector_type(2))) float v2f;
typedef __attribute__((ext_vector_type(8))) float v8f;

// ---------------------------------------------------------------------------
// Kernel 1: pooled[b,c] = mean over HW of x[b,c,:,:]
// ---------------------------------------------------------------------------
__global__ __launch_bounds__(128)
void pool_kernel(const float* __restrict__ x, float* __restrict__ pooled) {
    const int idx = blockIdx.x;                    // b*CIN + c
    const float* p = x + (size_t)idx * HW;
    float s = 0.f;
    for (int i = threadIdx.x; i < HW; i += 128) s += p[i];
    __shared__ float red[128];
    red[threadIdx.x] = s;
    __syncthreads();
    for (int off = 64; off > 0; off >>= 1) {
        if (threadIdx.x < off) red[threadIdx.x] += red[threadIdx.x + off];
        __syncthreads();
    }
    if (threadIdx.x == 0) pooled[idx] = red[0] * (1.0f / (float)HW);
}

// ---------------------------------------------------------------------------
// Kernel 2: attention trunk + heads. One block (256 thr) per batch sample.
// ---------------------------------------------------------------------------
__global__ __launch_bounds__(256)
void attn_kernel(const float* __restrict__ pooled,
                 const float* __restrict__ fc_w,
                 const float* __restrict__ abn_g, const float* __restrict__ abn_b,
                 const float* __restrict__ abn_m, const float* __restrict__ abn_v,
                 const float* __restrict__ ch_w,  const float* __restrict__ ch_b,
                 const float* __restrict__ fl_w,  const float* __restrict__ fl_b,
                 const float* __restrict__ kn_w,  const float* __restrict__ kn_b,
                 const float* __restrict__ bn_g,  const float* __restrict__ bn_b,
                 const float* __restrict__ bn_m,  const float* __restrict__ bn_v,
                 float* __restrict__ ch_att, float* __restrict__ scale,
                 float* __restrict__ kn_att, float* __restrict__ shift) {
    const int b = blockIdx.x;
    const int t = threadIdx.x;
    __shared__ float pool_s[CIN];
    __shared__ float hs[ATT];

    for (int c = t; c < CIN; c += 256) pool_s[c] = pooled[b * CIN + c];
    __syncthreads();

    if (t < ATT) {
        float acc = 0.f;
        for (int c = 0; c < CIN; ++c) acc += pool_s[c] * fc_w[t * CIN + c];
        acc = (acc - abn_m[t]) * (abn_g[t] * rsqrtf(abn_v[t] + EPS_)) + abn_b[t];
        hs[t] = fmaxf(acc, 0.f);
    }
    __syncthreads();

    for (int c = t; c < CIN; c += 256) {
        float acc = ch_b[c];
        for (int a = 0; a < ATT; ++a) acc += hs[a] * ch_w[c * ATT + a];
        ch_att[b * CIN + c] = 1.f / (1.f + expf(-acc));
    }
    for (int o = t; o < COUT; o += 256) {
        float acc = fl_b[o];
        for (int a = 0; a < ATT; ++a) acc += hs[a] * fl_w[o * ATT + a];
        float fl  = 1.f / (1.f + expf(-acc));
        float inv = bn_g[o] * rsqrtf(bn_v[o] + EPS_);
        scale[b * COUT + o] = fl * inv;
        if (b == 0) shift[o] = bn_b[o] - bn_m[o] * inv;
    }
    if (t == 0) {
        float lg[KNUM];
        float mx = -1e30f;
        for (int k = 0; k < KNUM; ++k) {
            float acc = kn_b[k];
            for (int a = 0; a < ATT; ++a) acc += hs[a] * kn_w[k * ATT + a];
            lg[k] = acc;
            mx = fmaxf(mx, acc);
        }
        float den = 0.f;
        for (int k = 0; k < KNUM; ++k) { lg[k] = expf(lg[k] - mx); den += lg[k]; }
        for (int k = 0; k < KNUM; ++k) kn_att[b * KNUM + k] = lg[k] / den;
    }
}

// ---------------------------------------------------------------------------
// Kernel 3: batched 1x1 conv as fp32 WMMA GEMM, double-buffered LDS pipeline.
//   y[b,o,p] = (sum_c W[o,c]*ch[c] * x[c,p]) * scale[b,o] + shift[o]
// Channel attention is folded into the aggregated W tile, so the X path is a
// pure global->LDS copy (branch-free: OOB pixel lanes are clamped to column 0;
// their accumulator columns are never stored).
// Block: 256 thr = 8 wave32s; block tile 64(M=cout) x 128(N=pixels).
// LDS layouts are K-contiguous with even stride 18 so every WMMA fragment is
// one aligned ds_load_b64 (conflict-free: 18*lane mod 64 distinct over 32 lanes).
// Pipeline: load chunk n+1 to regs -> WMMAs on chunk n -> spill regs to other
// buffer; global_prefetch for chunk n+2.
// ---------------------------------------------------------------------------
#define KSTRIDE 18   // even (b64-aligned fragments) and conflict-free

__global__ __launch_bounds__(256, 2)
void odconv_gemm_kernel(const float* __restrict__ x,
                        const float* __restrict__ weight,
                        const float* __restrict__ ch_att,
                        const float* __restrict__ kn_att,
                        const float* __restrict__ scale,
                        const float* __restrict__ shift,
                        float* __restrict__ y) {
    const int b     = blockIdx.z;
    const int oBase = blockIdx.x * 64;
    const int pBase = blockIdx.y * 128;
    const int t     = threadIdx.x;
    const int lane  = t & 31;
    const int wv    = t >> 5;
    const int waveM = (wv & 1) * 32;   // wave's M offset within block tile
    const int waveN = (wv >> 1) * 32;  // wave's N offset within block tile
    const int mr    = lane & 15;       // M/N index within a 16-tile
    const int hi    = lane >> 4;       // half-wave select

    __shared__ float Xs[2][128][KSTRIDE];  // [buf][pixel][k]
    __shared__ float Ws[2][64][KSTRIDE];   // [buf][o][k]

    const float* xb  = x      + (size_t)b * CIN * HW;
    const float* chb = ch_att + (size_t)b * CIN;
    const float* knb = kn_att + (size_t)b * KNUM;
    const float kn0 = knb[0], kn1 = knb[1], kn2 = knb[2], kn3 = knb[3];
    const size_t ks = (size_t)COUT * CIN;

    v8f acc[2][2] = {};

    const int col  = t & 127;          // pixel within X tile
    const int row0 = t >> 7;           // 0..1 (k phase for X staging)
    const int kW   = t & 15;           // k within W tile
    const int oW0  = t >> 4;           // 0..15 (o phase for W staging)
    const int p    = pBase + col;
    const bool pin = (p < HW);
    const int pc   = pin ? p : 0;      // clamp OOB lanes (columns never stored)

    // Thread's x base: channel (c) indexed as xt[c * HW], branch-free.
    const float* xt = xb + (size_t)row0 * HW + pc;
    // Thread's weight base for kernel 0: o indexed via (oBase+o)*CIN.
    const float* wb0 = weight + (size_t)oBase * CIN + kW;

    float xr[8], wr[4];

    // ---- prologue: stage chunk 0 ----
#pragma unroll
    for (int r = 0; r < 8; ++r) xr[r] = xt[(size_t)(2 * r) * HW];
    {
        const float chk = chb[kW];
#pragma unroll
        for (int r = 0; r < 4; ++r) {
            const float* wp = wb0 + (size_t)(oW0 + 16 * r) * CIN;
            wr[r] = (kn0 * wp[0] + kn1 * wp[ks] + kn2 * wp[2 * ks] + kn3 * wp[3 * ks]) * chk;
        }
    }
#pragma unroll
    for (int r = 0; r < 8; ++r) Xs[0][col][row0 + 2 * r] = xr[r];
#pragma unroll
    for (int r = 0; r < 4; ++r) Ws[0][oW0 + 16 * r][kW] = wr[r];

    const int NCHUNK = CIN / 16;       // 32
    for (int chunk = 0; chunk < NCHUNK; ++chunk) {
        const int cur = chunk & 1;
        __syncthreads();               // buffer `cur` ready for all waves

        // ---- issue global loads for chunk+1 (overlap with WMMAs below) ----
        const bool more = (chunk + 1 < NCHUNK);
        if (more) {
            const int c0n = (chunk + 1) * 16;
#pragma unroll
            for (int r = 0; r < 8; ++r) xr[r] = xt[(size_t)(c0n + 2 * r) * HW];
            const float chk = chb[c0n + kW];
#pragma unroll
            for (int r = 0; r < 4; ++r) {
                const float* wp = wb0 + (size_t)(oW0 + 16 * r) * CIN + c0n;
                wr[r] = (kn0 * wp[0] + kn1 * wp[ks] + kn2 * wp[2 * ks] + kn3 * wp[3 * ks]) * chk;
            }
        }
        // ---- L2 prefetch for chunk+2 (global_prefetch_b8) ----
        if (chunk + 2 < NCHUNK) {
            const int c2 = (chunk + 2) * 16;
            __builtin_prefetch(&xt[(size_t)c2 * HW], 0, 1);
            __builtin_prefetch(&wb0[(size_t)oW0 * CIN + c2], 0, 1);
        }

        // ---- 4 k-steps of V_WMMA_F32_16X16X4_F32 on buffer `cur` ----
#pragma unroll
        for (int kk = 0; kk < 16; kk += 4) {
            const int kb = kk + 2 * hi;   // K base for this half-wave (even)
            // A frag (16x4): lane L holds (M = L%16, K = 2*(L/16)+{0,1})
            const v2f a0  = *(const v2f*)&Ws[cur][waveM + mr][kb];
            const v2f a1  = *(const v2f*)&Ws[cur][waveM + 16 + mr][kb];
            // B frag (4x16): lane L holds (N = L%16, K = 2*(L/16)+{0,1})
            const v2f bx0 = *(const v2f*)&Xs[cur][waveN + mr][kb];
            const v2f bx1 = *(const v2f*)&Xs[cur][waveN + 16 + mr][kb];

            acc[0][0] = __builtin_amdgcn_wmma_f32_16x16x4_f32(
                false, a0, false, bx0, (short)0, acc[0][0], false, false);
            acc[0][1] = __builtin_amdgcn_wmma_f32_16x16x4_f32(
                false, a0, false, bx1, (short)0, acc[0][1], false, false);
            acc[1][0] = __builtin_amdgcn_wmma_f32_16x16x4_f32(
                false, a1, false, bx0, (short)0, acc[1][0], false, false);
            acc[1][1] = __builtin_amdgcn_wmma_f32_16x16x4_f32(
                false, a1, false, bx1, (short)0, acc[1][1], false, false);
        }

        // ---- spill staged chunk+1 into the other buffer ----
        if (more) {
            const int nxt = cur ^ 1;
#pragma unroll
            for (int r = 0; r < 8; ++r) Xs[nxt][col][row0 + 2 * r] = xr[r];
#pragma unroll
            for (int r = 0; r < 4; ++r) Ws[nxt][oW0 + 16 * r][kW] = wr[r];
        }
    }

    // ---- epilogue: D layout: VGPR r, lane<16 -> M=r, N=lane; lane>=16 -> M=r+8 ----
    const float* scb = scale + (size_t)b * COUT;
    float* yb = y + (size_t)b * COUT * HW;
#pragma unroll
    for (int mi = 0; mi < 2; ++mi) {
#pragma unroll
        for (int ni = 0; ni < 2; ++ni) {
            const int pp = pBase + waveN + ni * 16 + mr;
            if (pp < HW) {
#pragma unroll
                for (int r = 0; r < 8; ++r) {
                    const int o = oBase + waveM + mi * 16 + r + 8 * hi;
                    yb[(size_t)o * HW + pp] = acc[mi][ni][r] * scb[o] + shift[o];
                }
            }
        }
    }
}

// ---------------------------------------------------------------------------
// Launcher
// ---------------------------------------------------------------------------
extern "C" void kernel_launch(void* const* d_in, const int* in_sizes, int n_in,
                              void* d_out, int out_size, void* d_ws, size_t ws_size,
                              hipStream_t stream) {
    const float* x     = (const float*)d_in[0];
    const float* fc_w  = (const float*)d_in[1];
    const float* abn_g = (const float*)d_in[2];
    const float* abn_b = (const float*)d_in[3];
    const float* abn_m = (const float*)d_in[4];
    const float* abn_v = (const float*)d_in[5];
    const float* ch_w  = (const float*)d_in[6];
    const float* ch_b  = (const float*)d_in[7];
    const float* fl_w  = (const float*)d_in[8];
    const float* fl_b  = (const float*)d_in[9];
    const float* kn_w  = (const float*)d_in[10];
    const float* kn_b  = (const float*)d_in[11];
    const float* weight= (const float*)d_in[12];
    const float* bn_g  = (const float*)d_in[13];
    const float* bn_b  = (const float*)d_in[14];
    const float* bn_m  = (const float*)d_in[15];
    const float* bn_v  = (const float*)d_in[16];
    float* y = (float*)d_out;

    // Workspace layout (floats): pooled | ch_att | scale | kn_att | shift
    float* ws     = (float*)d_ws;
    float* pooled = ws;                       // B*CIN   = 16384
    float* ch_att = ws + 16384;               // B*CIN   = 16384
    float* scale  = ws + 32768;               // B*COUT  = 16384
    float* kn_att = ws + 49152;               // B*KNUM  = 128
    float* shift  = ws + 49280;               // COUT    = 512

    pool_kernel<<<B_ * CIN, 128, 0, stream>>>(x, pooled);

    attn_kernel<<<B_, 256, 0, stream>>>(pooled, fc_w, abn_g, abn_b, abn_m, abn_v,
                                        ch_w, ch_b, fl_w, fl_b, kn_w, kn_b,
                                        bn_g, bn_b, bn_m, bn_v,
                                        ch_att, scale, kn_att, shift);

    dim3 grid(COUT / 64, (HW + 127) / 128, B_);
    odconv_gemm_kernel<<<grid, 256, 0, stream>>>(x, weight, ch_att, kn_att,
                                                 scale, shift, y);
}